// div_metrics_84335977824352
// MI455X (gfx1250) — compile-verified
//
#include <hip/hip_runtime.h>

typedef float v2f __attribute__((ext_vector_type(2)));
typedef float v4f __attribute__((ext_vector_type(4)));
typedef float v8f __attribute__((ext_vector_type(8)));

// Per-element JSD contribution:
//   tw = (w>0 && m>0) ? w*(log2(w)-log2(m)) : 0
//   tp = (p>0 && m>0) ? p*(log2(p)-log2(m)) : 0
// (identity form avoids a v_rcp per element; 3x v_log_f32 only)
__device__ __forceinline__ float jsd_terms(float p, float w) {
    float m  = 0.5f * (w + p);
    float lm = __log2f(m);
    float tw = w * (__log2f(w) - lm);
    float tp = p * (__log2f(p) - lm);
    tw = (w > 0.0f && m > 0.0f) ? tw : 0.0f;   // v_cndmask kills 0*(-inf) NaN paths
    tp = (p > 0.0f && m > 0.0f) ? tp : 0.0f;
    return tw + tp;
}

__global__ void __launch_bounds__(256)
jsd_partial_kernel(const float* __restrict__ P, const float* __restrict__ W,
                   float* __restrict__ partial, long long n4) {
    const long long stride = (long long)gridDim.x * blockDim.x;
    long long idx = (long long)blockIdx.x * blockDim.x + threadIdx.x;

    const v4f* __restrict__ P4 = (const v4f*)P;
    const v4f* __restrict__ W4 = (const v4f*)W;

    // Two per-lane accumulators -> they become the lane's two B-operand VGPRs
    // for the WMMA wave reduction below.
    float acc0 = 0.0f, acc1 = 0.0f;

#pragma unroll 4
    for (; idx < n4; idx += stride) {
        // Single-use streaming data (268MB > 192MB L2): non-temporal b128 loads.
        v4f pv = __builtin_nontemporal_load(&P4[idx]);
        v4f wv = __builtin_nontemporal_load(&W4[idx]);
        acc0 += jsd_terms(pv.x, wv.x) + jsd_terms(pv.y, wv.y);
        acc1 += jsd_terms(pv.z, wv.z) + jsd_terms(pv.w, wv.w);
    }

    // ---- Wave32 reduction via V_WMMA_F32_16X16X4_F32 ----
    // A (16x4) = all ones  => D[m][n] = sum_k B[k][n]  (column sums of B).
    // B (4x16) holds all 64 per-lane accumulators of the wave (2 VGPRs/lane).
    // Every D row is the same 16-wide column-sum vector, so d[0] across a
    // 16-lane group enumerates all column sums; a 4-step xor-reduce over the
    // group gives the full wave total in every lane (layout-invariant).
    v2f a; a.x = 1.0f; a.y = 1.0f;
    v2f b; b.x = acc0; b.y = acc1;
    v8f c = {};
    v8f d = __builtin_amdgcn_wmma_f32_16x16x4_f32(
        /*neg_a=*/false, a, /*neg_b=*/false, b,
        /*c_mod=*/(short)0, c, /*reuse_a=*/false, /*reuse_b=*/false);

    float wsum = d[0];
    wsum += __shfl_xor(wsum, 1, 32);
    wsum += __shfl_xor(wsum, 2, 32);
    wsum += __shfl_xor(wsum, 4, 32);
    wsum += __shfl_xor(wsum, 8, 32);
    // lanes 0..15 and 16..31 each now hold the identical wave total.

    __shared__ float lds[8];
    const int lane = threadIdx.x & 31;
    const int wave = threadIdx.x >> 5;
    if (lane == 0) lds[wave] = wsum;
    __syncthreads();
    if (threadIdx.x == 0) {
        float s = 0.0f;
#pragma unroll
        for (int i = 0; i < 8; ++i) s += lds[i];
        partial[blockIdx.x] = s;
    }
}

__global__ void __launch_bounds__(256)
jsd_final_kernel(const float* __restrict__ partial, float* __restrict__ out,
                 int nparts, float scale) {
    __shared__ double lds[256];
    double s = 0.0;
    for (int i = threadIdx.x; i < nparts; i += 256) s += (double)partial[i];
    lds[threadIdx.x] = s;
    __syncthreads();
#pragma unroll
    for (int off = 128; off >= 1; off >>= 1) {
        if ((int)threadIdx.x < off) lds[threadIdx.x] += lds[threadIdx.x + off];
        __syncthreads();
    }
    if (threadIdx.x == 0) out[0] = (float)(lds[0] * (double)scale);
}

extern "C" void kernel_launch(void* const* d_in, const int* in_sizes, int n_in,
                              void* d_out, int out_size, void* d_ws, size_t ws_size,
                              hipStream_t stream) {
    const float* P = (const float*)d_in[0];
    const float* W = (const float*)d_in[1];
    float* partial = (float*)d_ws;

    const long long n  = (long long)in_sizes[0];   // 8192*4096
    const long long n4 = n >> 2;                   // float4 count
    const int blocks = 2048;                       // 2048*256 lanes -> 16 iters/lane

    // result = (S_w + S_p) / (2 * vn), vn = rows = n / 4096
    const float scale = 1.0f / (2.0f * (float)(n / 4096));

    jsd_partial_kernel<<<blocks, 256, 0, stream>>>(P, W, partial, n4);
    jsd_final_kernel<<<1, 256, 0, stream>>>(partial, (float*)d_out, blocks, scale);
}